// RBFLayer_51058571215604
// MI455X (gfx1250) — compile-verified
//
#include <hip/hip_runtime.h>

typedef float v2f __attribute__((ext_vector_type(2)));
typedef float v8f __attribute__((ext_vector_type(8)));

#define BATCH 8192
#define UNITS 2048
#define DIM   32
#define NT    4   // 16x16 output tiles per wave along the UNITS dimension

// Each wave computes a 16x(16*NT) strip of out[B, UNITS].
// sq_dist[b,u] = |x_b|^2 + |c_u|^2 - 2 * dot(x_b, c_u); dot via 8 chained
// V_WMMA_F32_16X16X4_F32 (K = 32 = 8 * 4), all in fp32 like the reference.
__global__ __launch_bounds__(256) void rbf_wmma_kernel(
    const float* __restrict__ x,        // [BATCH, DIM]
    const float* __restrict__ centers,  // [UNITS, DIM]
    const float* __restrict__ gamma,    // [1]
    float* __restrict__ out)            // [BATCH, UNITS]
{
    const int lane = threadIdx.x & 31;
    const int gw   = blockIdx.x * 8 + (threadIdx.x >> 5);   // global wave id

    const int wavesPerRow = UNITS / (16 * NT);              // 32
    const int b0 = (gw / wavesPerRow) * 16;                 // tile row base
    const int u0 = (gw % wavesPerRow) * (16 * NT);          // tile col base

    const int g  = lane >> 4;   // lane group (0: lanes 0-15, 1: lanes 16-31)
    const int mn = lane & 15;   // A row index / B column index held by lane

    const float gam = gamma[0];

    // ---- A fragments: a[i] holds x[b0+mn][4i + 2g + {0,1}] (one float2 load)
    v2f a[8];
    float xn_part = 0.f;
    const float* xrow = x + (b0 + mn) * DIM + 2 * g;
#pragma unroll
    for (int i = 0; i < 8; ++i) {
        a[i] = *(const v2f*)(xrow + 4 * i);
        xn_part += a[i].x * a[i].x + a[i].y * a[i].y;
    }
    // lane holds half of row mn's elements; lane^16 holds the other half
    const float xn = xn_part + __shfl_xor(xn_part, 16, 32); // |x_{b0+mn}|^2

    v8f   acc[NT];
    float cn[NT];

#pragma unroll
    for (int t = 0; t < NT; ++t) {
        // ---- B fragments: b[i] holds centers[u0+16t+mn][4i + 2g + {0,1}]
        const float* crow = centers + (u0 + t * 16 + mn) * DIM + 2 * g;
        v2f bfrag[8];
        float cn_part = 0.f;
#pragma unroll
        for (int i = 0; i < 8; ++i) {
            bfrag[i] = *(const v2f*)(crow + 4 * i);
            cn_part += bfrag[i].x * bfrag[i].x + bfrag[i].y * bfrag[i].y;
        }
        cn[t] = cn_part + __shfl_xor(cn_part, 16, 32);      // |c_{u0+16t+mn}|^2

        v8f c = {0.f, 0.f, 0.f, 0.f, 0.f, 0.f, 0.f, 0.f};
#pragma unroll
        for (int i = 0; i < 8; ++i) {
            // D = A(16x4) * B(4x16) + C, fp32 in / fp32 accumulate
            c = __builtin_amdgcn_wmma_f32_16x16x4_f32(
                    /*neg_a=*/false, a[i],
                    /*neg_b=*/false, bfrag[i],
                    /*c_mod=*/(short)0, c,
                    /*reuse_a=*/false, /*reuse_b=*/false);
        }
        acc[t] = c;
    }

    // ---- Epilogue. C/D layout: vgpr r, lane -> (M = r + 8g, N = mn).
    // c_norm for column N=mn is already resident in this lane; x_norm for
    // row r+8g is fetched with a per-lane-source shuffle (ds_bpermute).
#pragma unroll
    for (int r = 0; r < 8; ++r) {
        const int   m   = r + 8 * g;
        const float xnr = __shfl(xn, m, 32);    // |x_{b0+m}|^2
        float* orow = out + (b0 + m) * UNITS + u0 + mn;
#pragma unroll
        for (int t = 0; t < NT; ++t) {
            float sq = xnr + cn[t] - 2.f * acc[t][r];
            sq = fmaxf(sq, 0.f);                // guard tiny negatives from expansion
            // streamed 64 MB output, never re-read on device -> non-temporal
            __builtin_nontemporal_store(__expf(-gam * sq), orow + t * 16);
        }
    }
}

extern "C" void kernel_launch(void* const* d_in, const int* in_sizes, int n_in,
                              void* d_out, int out_size, void* d_ws, size_t ws_size,
                              hipStream_t stream)
{
    const float* x       = (const float*)d_in[0];
    const float* centers = (const float*)d_in[1];
    const float* gamma   = (const float*)d_in[2];
    float*       out     = (float*)d_out;

    // waves = (8192/16) * (2048/64) = 16384 ; 8 waves per 256-thread block
    const int totalWaves = (BATCH / 16) * (UNITS / (16 * NT));
    dim3 grid(totalWaves / 8), block(256);
    hipLaunchKernelGGL(rbf_wmma_kernel, grid, block, 0, stream,
                       x, centers, gamma, out);
}